// GraphBinaryClassificationModel_28509992910994
// MI455X (gfx1250) — compile-verified
//
#include <hip/hip_runtime.h>
#include <math.h>

typedef __attribute__((ext_vector_type(2))) float v2f;
typedef __attribute__((ext_vector_type(8))) float v8f;

#define NNODES 100000
#define NEDGES 1600000
#define NGRAPHS 256

// ---------------------------------------------------------------------------
// Zero a float buffer
__global__ void zero_kernel(float* __restrict__ p, long long n) {
    long long i = (long long)blockIdx.x * blockDim.x + threadIdx.x;
    if (i < n) p[i] = 0.0f;
}

// Pad x [100000,29] -> h0 [100000,32] (cols 29..31 = 0)
__global__ void pad_x_kernel(const float* __restrict__ x, float* __restrict__ h0) {
    int idx = blockIdx.x * blockDim.x + threadIdx.x;
    if (idx >= NNODES * 32) return;
    int node = idx >> 5, f = idx & 31;
    h0[idx] = (f < 29) ? x[node * 29 + f] : 0.0f;
}

// Pad w1_rel / w1_root [29,256] -> [32,256] (rows 29..31 = 0)
__global__ void pad_w1_kernel(const float* __restrict__ wr, const float* __restrict__ wo,
                              float* __restrict__ wrp, float* __restrict__ wop) {
    int idx = blockIdx.x * blockDim.x + threadIdx.x;
    if (idx >= 32 * 256) return;
    int k = idx >> 8, n = idx & 255;
    wrp[idx] = (k < 29) ? wr[k * 256 + n] : 0.0f;
    wop[idx] = (k < 29) ? wo[k * 256 + n] : 0.0f;
}

// Fold bias + (optional) BN into per-column affine: out = acc*alpha + beta
__global__ void make_affine_kernel(const float* __restrict__ b_rel,
                                   const float* __restrict__ g, const float* __restrict__ bb,
                                   const float* __restrict__ m, const float* __restrict__ v,
                                   float* __restrict__ alpha, float* __restrict__ beta, int n) {
    int i = blockIdx.x * blockDim.x + threadIdx.x;
    if (i >= n) return;
    if (g != nullptr) {
        float s = g[i] * rsqrtf(v[i] + 1e-5f);
        alpha[i] = s;
        beta[i]  = (b_rel[i] - m[i]) * s + bb[i];
    } else {
        alpha[i] = 1.0f;
        beta[i]  = b_rel[i];
    }
}

// Edge scatter, 4 features per thread: agg[dst] += ew * y[src]
// F = 1<<lf wide (row-major [N,F], F multiple of 4). One b128 gather + 4 atomics.
__global__ void scatter4_kernel(const float* __restrict__ y,
                                const long long* __restrict__ src,
                                const long long* __restrict__ dst,
                                const float* __restrict__ ew,
                                float* __restrict__ agg, int lf, long long total4) {
    long long idx = (long long)blockIdx.x * blockDim.x + threadIdx.x;
    if (idx >= total4) return;
    const int lf4 = lf - 2;
    long long e = idx >> lf4;
    int f = (int)(idx & ((1LL << lf4) - 1)) << 2;
    long long s = src[e], d = dst[e];
    float w = ew[e];
    float4 yv = *reinterpret_cast<const float4*>(y + (s << lf) + f);
    float* ap = agg + (d << lf) + f;
    atomicAdd(ap + 0, w * yv.x);
    atomicAdd(ap + 1, w * yv.y);
    atomicAdd(ap + 2, w * yv.z);
    atomicAdd(ap + 3, w * yv.w);
}

// ---------------------------------------------------------------------------
// WMMA fp32 GEMM: D = epilogue( A1@B1 [+ A2@B2] [+ Cadd] ) with per-col affine
// Each wave computes one 16x16 tile via V_WMMA_F32_16X16X4_F32.
// A row-major [M,K] (lda=K), B row-major [K,N].
__device__ __forceinline__ v8f wmma_seg(const float* __restrict__ A,
                                        const float* __restrict__ B,
                                        int K, int N, int row0, int col0,
                                        int l16, int koff, v8f acc) {
    const float* arow = A + (size_t)(row0 + l16) * K + koff;
    const float* bcol = B + (size_t)koff * N + col0 + l16;
    for (int k = 0; k < K; k += 4) {
        float2 av = *reinterpret_cast<const float2*>(arow + k);
        v2f a; a.x = av.x; a.y = av.y;
        const float* bp = bcol + (size_t)k * N;
        v2f b; b.x = bp[0]; b.y = bp[N];
        acc = __builtin_amdgcn_wmma_f32_16x16x4_f32(false, a, false, b,
                                                    (short)0, acc, false, false);
    }
    return acc;
}

__global__ void wmma_gemm_kernel(const float* __restrict__ A1, int K1,
                                 const float* __restrict__ B1,
                                 const float* __restrict__ A2, int K2,
                                 const float* __restrict__ B2,
                                 const float* __restrict__ Cadd,
                                 const float* __restrict__ alpha,
                                 const float* __restrict__ beta,
                                 float* __restrict__ D,
                                 int M, int N, int do_relu) {
    const int lane = threadIdx.x & 31;
    const int wib  = threadIdx.x >> 5;
    const int ntn  = N >> 4;
    const long long wid   = (long long)blockIdx.x * 8 + wib;
    const long long total = (long long)(M >> 4) * ntn;
    if (wid >= total) return;                 // wave-uniform: EXEC all-ones inside
    const int tn = (int)(wid % ntn);
    const int tm = (int)(wid / ntn);
    const int row0 = tm << 4, col0 = tn << 4;
    const int hi   = lane >> 4;               // lane half select
    const int l16  = lane & 15;
    const int koff = hi ? 2 : 0;
    const int roff = hi ? 8 : 0;

    v8f acc = {};
    if (Cadd) {
#pragma unroll
        for (int v = 0; v < 8; ++v)
            acc[v] = Cadd[(size_t)(row0 + v + roff) * N + col0 + l16];
    }

    acc = wmma_seg(A1, B1, K1, N, row0, col0, l16, koff, acc);
    if (K2 > 0)
        acc = wmma_seg(A2, B2, K2, N, row0, col0, l16, koff, acc);

    const int col = col0 + l16;
    const float al = alpha ? alpha[col] : 1.0f;
    const float be = alpha ? beta[col]  : 0.0f;
#pragma unroll
    for (int v = 0; v < 8; ++v) {
        int row = row0 + v + roff;
        float val = acc[v] * al + be;
        if (do_relu) val = fmaxf(val, 0.0f);
        D[(size_t)row * N + col] = val;
    }
}

// ---------------------------------------------------------------------------
// Global mean pool (4 features per thread) + FC + sigmoid
__global__ void pool4_kernel(const float* __restrict__ h,
                             const long long* __restrict__ batch,
                             float* __restrict__ pool) {
    int idx = blockIdx.x * blockDim.x + threadIdx.x;   // NNODES * 8 threads
    if (idx >= NNODES * 8) return;
    int node = idx >> 3, f = (idx & 7) << 2;
    float4 hv = *reinterpret_cast<const float4*>(h + node * 32 + f);
    float* pp = pool + (int)batch[node] * 32 + f;
    atomicAdd(pp + 0, hv.x);
    atomicAdd(pp + 1, hv.y);
    atomicAdd(pp + 2, hv.z);
    atomicAdd(pp + 3, hv.w);
}

__global__ void count_kernel(const long long* __restrict__ batch, float* __restrict__ cnt) {
    int i = blockIdx.x * blockDim.x + threadIdx.x;
    if (i >= NNODES) return;
    atomicAdd(&cnt[(int)batch[i]], 1.0f);
}

__global__ void final_kernel(const float* __restrict__ pool, const float* __restrict__ cnt,
                             const float* __restrict__ fc_w, const float* __restrict__ fc_b,
                             float* __restrict__ out) {
    int g = threadIdx.x;
    if (g >= NGRAPHS) return;
    float c = fmaxf(cnt[g], 1.0f);
    float s = 0.0f;
#pragma unroll
    for (int f = 0; f < 32; ++f) s += (pool[g * 32 + f] / c) * fc_w[f];
    s += fc_b[0];
    out[g] = 1.0f / (1.0f + expf(-s));
}

// ---------------------------------------------------------------------------
static inline int cdiv(long long a, long long b) { return (int)((a + b - 1) / b); }

extern "C" void kernel_launch(void* const* d_in, const int* in_sizes, int n_in,
                              void* d_out, int out_size, void* d_ws, size_t ws_size,
                              hipStream_t stream) {
    (void)in_sizes; (void)n_in; (void)out_size; (void)ws_size;
    const float*     x     = (const float*)d_in[0];
    const long long* eidx  = (const long long*)d_in[1];
    const float*     ew    = (const float*)d_in[2];
    const long long* batch = (const long long*)d_in[3];
    const float *w1_rel=(const float*)d_in[4],  *b1_rel=(const float*)d_in[5],  *w1_root=(const float*)d_in[6];
    const float *w2_rel=(const float*)d_in[7],  *b2_rel=(const float*)d_in[8],  *w2_root=(const float*)d_in[9];
    const float *w3_rel=(const float*)d_in[10], *b3_rel=(const float*)d_in[11], *w3_root=(const float*)d_in[12];
    const float *w4_rel=(const float*)d_in[13], *b4_rel=(const float*)d_in[14], *w4_root=(const float*)d_in[15];
    const float *bn1_g=(const float*)d_in[16], *bn1_b=(const float*)d_in[17], *bn1_m=(const float*)d_in[18], *bn1_v=(const float*)d_in[19];
    const float *bn2_g=(const float*)d_in[20], *bn2_b=(const float*)d_in[21], *bn2_m=(const float*)d_in[22], *bn2_v=(const float*)d_in[23];
    const float *bn3_g=(const float*)d_in[24], *bn3_b=(const float*)d_in[25], *bn3_m=(const float*)d_in[26], *bn3_v=(const float*)d_in[27];
    const float *fc_w=(const float*)d_in[28], *fc_b=(const float*)d_in[29];
    float* out = (float*)d_out;

    const long long* src = eidx;           // edge_index[0]
    const long long* dst = eidx + NEDGES;  // edge_index[1]

    // Workspace layout (floats), with buffer reuse across layers
    float* ws = (float*)d_ws;
    size_t o = 0;
    float* R0 = ws + o; o += (size_t)NNODES * 256;  // h1 -> y3 -> y4
    float* R1 = ws + o; o += (size_t)NNODES * 128;  // y2 -> agg3 -> agg4
    float* R2 = ws + o; o += (size_t)NNODES * 128;  // agg2 -> h3
    float* R3 = ws + o; o += (size_t)NNODES * 128;  // h2 -> h4
    float* h0   = ws + o; o += (size_t)NNODES * 32;
    float* agg1 = ws + o; o += (size_t)NNODES * 32;
    float* w1p_rel  = ws + o; o += 32 * 256;
    float* w1p_root = ws + o; o += 32 * 256;
    float* al1 = ws + o; o += 256; float* be1 = ws + o; o += 256;
    float* al2 = ws + o; o += 128; float* be2 = ws + o; o += 128;
    float* al3 = ws + o; o += 64;  float* be3 = ws + o; o += 64;
    float* al4 = ws + o; o += 32;  float* be4 = ws + o; o += 32;
    float* pool = ws + o; o += NGRAPHS * 32;
    float* cnt  = ws + o; o += NGRAPHS;

    const int T = 256;
    // --- prep: padding + folded affines
    pad_x_kernel<<<cdiv((long long)NNODES * 32, T), T, 0, stream>>>(x, h0);
    pad_w1_kernel<<<cdiv(32 * 256, T), T, 0, stream>>>(w1_rel, w1_root, w1p_rel, w1p_root);
    make_affine_kernel<<<1, 256, 0, stream>>>(b1_rel, nullptr, nullptr, nullptr, nullptr, al1, be1, 256);
    make_affine_kernel<<<1, 128, 0, stream>>>(b2_rel, bn1_g, bn1_b, bn1_m, bn1_v, al2, be2, 128);
    make_affine_kernel<<<1, 64,  0, stream>>>(b3_rel, bn2_g, bn2_b, bn2_m, bn2_v, al3, be3, 64);
    make_affine_kernel<<<1, 32,  0, stream>>>(b4_rel, bn3_g, bn3_b, bn3_m, bn3_v, al4, be4, 32);

    // --- Layer 1 (29->256): aggregate in (padded) input dim, then dual GEMM
    zero_kernel<<<cdiv((long long)NNODES * 32, T), T, 0, stream>>>(agg1, (long long)NNODES * 32);
    scatter4_kernel<<<cdiv((long long)NEDGES * 8, T), T, 0, stream>>>(h0, src, dst, ew, agg1, 5, (long long)NEDGES * 8);
    {   // h1 = relu(agg1@w1p_rel + h0@w1p_root + b1)  [M=100000,N=256]
        long long tiles = (NNODES / 16) * (256 / 16);
        wmma_gemm_kernel<<<cdiv(tiles, 8), T, 0, stream>>>(agg1, 32, w1p_rel, h0, 32, w1p_root,
                                                           nullptr, al1, be1, R0, NNODES, 256, 1);
    }
    // --- Layer 2 (256->128): pre-multiply by w_rel, aggregate at 128
    {   // y2 = h1 @ w2_rel
        long long tiles = (NNODES / 16) * (128 / 16);
        wmma_gemm_kernel<<<cdiv(tiles, 8), T, 0, stream>>>(R0, 256, w2_rel, nullptr, 0, nullptr,
                                                           nullptr, nullptr, nullptr, R1, NNODES, 128, 0);
        zero_kernel<<<cdiv((long long)NNODES * 128, T), T, 0, stream>>>(R2, (long long)NNODES * 128);
        scatter4_kernel<<<cdiv((long long)NEDGES * 32, T), T, 0, stream>>>(R1, src, dst, ew, R2, 7, (long long)NEDGES * 32);
        // h2 = relu(bn1(h1@w2_root + agg2 + b2))
        wmma_gemm_kernel<<<cdiv(tiles, 8), T, 0, stream>>>(R0, 256, w2_root, nullptr, 0, nullptr,
                                                           R2, al2, be2, R3, NNODES, 128, 1);
    }
    // --- Layer 3 (128->64)
    {
        long long tiles = (NNODES / 16) * (64 / 16);
        wmma_gemm_kernel<<<cdiv(tiles, 8), T, 0, stream>>>(R3, 128, w3_rel, nullptr, 0, nullptr,
                                                           nullptr, nullptr, nullptr, R0, NNODES, 64, 0); // y3
        zero_kernel<<<cdiv((long long)NNODES * 64, T), T, 0, stream>>>(R1, (long long)NNODES * 64);
        scatter4_kernel<<<cdiv((long long)NEDGES * 16, T), T, 0, stream>>>(R0, src, dst, ew, R1, 6, (long long)NEDGES * 16);
        wmma_gemm_kernel<<<cdiv(tiles, 8), T, 0, stream>>>(R3, 128, w3_root, nullptr, 0, nullptr,
                                                           R1, al3, be3, R2, NNODES, 64, 1);            // h3
    }
    // --- Layer 4 (64->32)
    {
        long long tiles = (NNODES / 16) * (32 / 16);
        wmma_gemm_kernel<<<cdiv(tiles, 8), T, 0, stream>>>(R2, 64, w4_rel, nullptr, 0, nullptr,
                                                           nullptr, nullptr, nullptr, R0, NNODES, 32, 0); // y4
        zero_kernel<<<cdiv((long long)NNODES * 32, T), T, 0, stream>>>(R1, (long long)NNODES * 32);
        scatter4_kernel<<<cdiv((long long)NEDGES * 8, T), T, 0, stream>>>(R0, src, dst, ew, R1, 5, (long long)NEDGES * 8);
        wmma_gemm_kernel<<<cdiv(tiles, 8), T, 0, stream>>>(R2, 64, w4_root, nullptr, 0, nullptr,
                                                           R1, al4, be4, R3, NNODES, 32, 1);            // h4
    }
    // --- Mean pool + FC + sigmoid
    zero_kernel<<<cdiv(NGRAPHS * 32 + NGRAPHS, T), T, 0, stream>>>(pool, NGRAPHS * 32 + NGRAPHS);
    pool4_kernel<<<cdiv((long long)NNODES * 8, T), T, 0, stream>>>(R3, batch, pool);
    count_kernel<<<cdiv(NNODES, T), T, 0, stream>>>(batch, cnt);
    final_kernel<<<1, 256, 0, stream>>>(pool, cnt, fc_w, fc_b, out);
}